// CompPCFG_20658792694660
// MI455X (gfx1250) — compile-verified
//
#include <hip/hip_runtime.h>
#include <hip/hip_bf16.h>
#include <math.h>

typedef __attribute__((ext_vector_type(2))) float v2f;
typedef __attribute__((ext_vector_type(8))) float v8f;

constexpr int VOCABN = 10000;
constexpr int WDIMN  = 512;
constexpr int HDIMN  = 512;
constexpr int ZDIMN  = 64;
constexpr int SDIMN  = 256;
constexpr int TSN    = 60;
constexpr int NTSN   = 30;
constexpr int SALLN  = 90;   // TSN + NTSN
constexpr int BBN    = 4;
constexpr int NNN    = 24;
constexpr float NEGV = -1000000000.0f;

__device__ __forceinline__ float sigf(float x) { return 1.0f / (1.0f + expf(-x)); }

// ---------------------------------------------------------------------------
// Generic batched WMMA f32 GEMM: C[M,N] = A[M,K] @ W[N,K]^T + bias, opt ReLU,
// opt residual add (after ReLU). One wave (32 threads) per 32x32 output tile.
// Uses V_WMMA_F32_16X16X4_F32 (exact f32 matrix core path on gfx1250).
// ---------------------------------------------------------------------------
__device__ __forceinline__ void store_tile(const v8f& acc, float* C, const float* bias,
                                           const float* resid, int M, int N,
                                           int rb, int cb, int lane, int relu) {
  int col = cb + (lane & 15);
  if (col >= N) return;
  float bv = bias ? bias[col] : 0.0f;
  int rbase = rb + ((lane >> 4) << 3);
#pragma unroll
  for (int r = 0; r < 8; ++r) {
    int row = rbase + r;
    if (row < M) {
      float v = acc[r] + bv;
      if (relu) v = fmaxf(v, 0.0f);
      if (resid) v += resid[(long)row * N + col];
      C[(long)row * N + col] = v;
    }
  }
}

__global__ __launch_bounds__(32)
void wmma_gemm_kernel(const float* __restrict__ A, const float* __restrict__ Wt,
                      const float* __restrict__ bias, const float* __restrict__ resid,
                      float* __restrict__ C, int M, int N, int K,
                      long sA, long sW, long sC, long sR, int relu) {
  int bz = blockIdx.z;
  A += bz * sA; Wt += bz * sW; C += bz * sC;
  if (resid) resid += bz * sR;
  int tn = blockIdx.x * 32;
  int tm = blockIdx.y * 32;
  int l  = threadIdx.x;
  int lm = l & 15;
  int lk = (l >> 4) << 1;                 // K sub-offset {0,2} per ISA A/B f32 layout
  int rA0 = min(tm + lm, M - 1);
  int rA1 = min(tm + 16 + lm, M - 1);
  int nB0 = min(tn + lm, N - 1);
  int nB1 = min(tn + 16 + lm, N - 1);
  const float* pa0 = A  + (long)rA0 * K + lk;
  const float* pa1 = A  + (long)rA1 * K + lk;
  const float* pb0 = Wt + (long)nB0 * K + lk;
  const float* pb1 = Wt + (long)nB1 * K + lk;
  v8f acc00 = {}, acc01 = {}, acc10 = {}, acc11 = {};
  for (int kb = 0; kb < K; kb += 4) {
    v2f a0 = *(const v2f*)(pa0 + kb);
    v2f a1 = *(const v2f*)(pa1 + kb);
    v2f b0 = *(const v2f*)(pb0 + kb);
    v2f b1 = *(const v2f*)(pb1 + kb);
    acc00 = __builtin_amdgcn_wmma_f32_16x16x4_f32(false, a0, false, b0, (short)0, acc00, false, false);
    acc01 = __builtin_amdgcn_wmma_f32_16x16x4_f32(false, a0, false, b1, (short)0, acc01, false, false);
    acc10 = __builtin_amdgcn_wmma_f32_16x16x4_f32(false, a1, false, b0, (short)0, acc10, false, false);
    acc11 = __builtin_amdgcn_wmma_f32_16x16x4_f32(false, a1, false, b1, (short)0, acc11, false, false);
  }
  store_tile(acc00, C, bias, resid, M, N, tm,      tn,      l, relu);
  store_tile(acc01, C, bias, resid, M, N, tm,      tn + 16, l, relu);
  store_tile(acc10, C, bias, resid, M, N, tm + 16, tn,      l, relu);
  store_tile(acc11, C, bias, resid, M, N, tm + 16, tn + 16, l, relu);
}

// ---------------------------------------------------------------------------
// Small helper kernels
// ---------------------------------------------------------------------------
__global__ void fill_kernel(float* p, long n, float v) {
  long i = (long)blockIdx.x * 256 + threadIdx.x;
  if (i < n) p[i] = v;
}

__global__ void emb_gather_kernel(const int* __restrict__ x, const float* __restrict__ enc_emb,
                                  float* __restrict__ emb) {
  int idx = blockIdx.x * 256 + threadIdx.x;
  if (idx >= BBN * NNN * WDIMN) return;
  int row = idx / WDIMN;
  int col = idx - row * WDIMN;
  int tok = x[row];
  emb[idx] = enc_emb[(long)tok * WDIMN + col];
}

// concat(state_emb[st], z[b]) -> rows = BBN*nst, cols = SDIM+ZDIM
__global__ void cat_in_kernel(const float* __restrict__ st_emb, const float* __restrict__ z,
                              float* __restrict__ out, int nst) {
  int cols = SDIMN + ZDIMN;
  int idx = blockIdx.x * 256 + threadIdx.x;
  if (idx >= BBN * nst * cols) return;
  int row = idx / cols;
  int col = idx - row * cols;
  int b = row / nst;
  int st = row - b * nst;
  out[idx] = (col < SDIMN) ? st_emb[st * SDIMN + col] : z[b * ZDIMN + (col - SDIMN)];
}

// One block per direction; full LSTM step (g = pre[t] + h@Whh^T + bhh, gates).
__global__ __launch_bounds__(256)
void lstm_step_kernel(const float* __restrict__ preF, const float* __restrict__ preB,
                      const float* __restrict__ WhhF, const float* __restrict__ WhhB,
                      const float* __restrict__ bhhF, const float* __restrict__ bhhB,
                      float* __restrict__ hF, float* __restrict__ cF,
                      float* __restrict__ hB, float* __restrict__ cB,
                      float* __restrict__ hsF, float* __restrict__ hsB, int t) {
  __shared__ float hsh[BBN * HDIMN];
  __shared__ float g[BBN * 4 * HDIMN];
  int dir = blockIdx.x;
  const float* pre = dir ? preB : preF;
  const float* Whh = dir ? WhhB : WhhF;
  const float* bhh = dir ? bhhB : bhhF;
  float* h = dir ? hB : hF;
  float* c = dir ? cB : cF;
  float* hs = dir ? hsB : hsF;
  int tt = dir ? (NNN - 1 - t) : t;
  int tid = threadIdx.x;
  for (int i = tid; i < BBN * HDIMN; i += 256) hsh[i] = h[i];
  __syncthreads();
  for (int idx = tid; idx < BBN * 4 * HDIMN; idx += 256) {
    int bi = idx >> 11;                    // /2048
    int j  = idx & (4 * HDIMN - 1);
    const float* wr = Whh + (long)j * HDIMN;
    const float* hr = hsh + bi * HDIMN;
    float acc = bhh[j] + pre[((long)(bi * NNN + tt)) * (4 * HDIMN) + j];
    for (int k = 0; k < HDIMN; ++k) acc += hr[k] * wr[k];
    g[idx] = acc;
  }
  __syncthreads();
  for (int idx = tid; idx < BBN * HDIMN; idx += 256) {
    int bi = idx >> 9;                     // /512
    int k  = idx & (HDIMN - 1);
    const float* gb = g + bi * 4 * HDIMN;
    float gi = gb[k], gf = gb[HDIMN + k], gg = gb[2 * HDIMN + k], go = gb[3 * HDIMN + k];
    float cc = sigf(gf) * c[idx] + sigf(gi) * tanhf(gg);
    float hh = sigf(go) * tanhf(cc);
    c[idx] = cc; h[idx] = hh;
    hs[((long)(bi * NNN + tt)) * HDIMN + k] = hh;
  }
}

__global__ __launch_bounds__(1024)
void hmax_kernel(const float* __restrict__ hsF, const float* __restrict__ hsB,
                 float* __restrict__ hmax) {
  int b = blockIdx.x, j = threadIdx.x;
  float mx = -INFINITY;
  for (int t = 0; t < NNN; ++t) {
    float v = (j < HDIMN) ? hsF[((long)(b * NNN + t)) * HDIMN + j]
                          : hsB[((long)(b * NNN + t)) * HDIMN + (j - HDIMN)];
    mx = fmaxf(mx, v);
  }
  hmax[b * 2 * HDIMN + j] = mx;
}

__global__ __launch_bounds__(128)
void ep_kernel(const float* __restrict__ hmax, const float* __restrict__ ep_W,
               const float* __restrict__ ep_b, const float* __restrict__ eps,
               float* __restrict__ z, float* __restrict__ kl) {
  __shared__ float epl[2 * ZDIMN];
  __shared__ float red[64];
  int b = blockIdx.x, tid = threadIdx.x;
  {
    const float* hr = hmax + b * 2 * HDIMN;
    const float* wr = ep_W + (long)tid * 2 * HDIMN;
    float acc = ep_b[tid];
    for (int k = 0; k < 2 * HDIMN; ++k) acc += hr[k] * wr[k];
    epl[tid] = acc;
  }
  __syncthreads();
  if (tid < ZDIMN) {
    float mean = epl[tid], lv = epl[ZDIMN + tid];
    z[b * ZDIMN + tid] = expf(0.5f * lv) * eps[b * ZDIMN + tid] + mean;
    red[tid] = -0.5f * (lv - mean * mean - expf(lv) + 1.0f);
  }
  __syncthreads();
  for (int s = 32; s > 0; s >>= 1) { if (tid < s) red[tid] += red[tid + s]; __syncthreads(); }
  if (tid == 0) kl[b] = red[0];
}

__global__ __launch_bounds__(256)
void root_mlp_kernel(const float* __restrict__ root_emb, const float* __restrict__ z,
                     const float* __restrict__ W0, const float* __restrict__ b0,
                     const float* __restrict__ W11, const float* __restrict__ b11,
                     const float* __restrict__ W12, const float* __restrict__ b12,
                     const float* __restrict__ W21, const float* __restrict__ b21,
                     const float* __restrict__ W22, const float* __restrict__ b22,
                     const float* __restrict__ Wo, const float* __restrict__ bo,
                     float* __restrict__ root_scores) {
  __shared__ float xin[SDIMN + ZDIMN];
  __shared__ float Ah[SDIMN], Bh[SDIMN], Ch[SDIMN];
  __shared__ float ov[NTSN];
  __shared__ float lse1;
  int b = blockIdx.x, tid = threadIdx.x;
  xin[tid] = root_emb[tid];
  if (tid < ZDIMN) xin[SDIMN + tid] = z[b * ZDIMN + tid];
  __syncthreads();
  { const float* wr = W0 + (long)tid * (SDIMN + ZDIMN); float a = b0[tid];
    for (int k = 0; k < SDIMN + ZDIMN; ++k) a += xin[k] * wr[k]; Ah[tid] = a; }
  __syncthreads();
  { const float* wr = W11 + (long)tid * SDIMN; float a = b11[tid];
    for (int k = 0; k < SDIMN; ++k) a += Ah[k] * wr[k]; Bh[tid] = fmaxf(a, 0.0f); }
  __syncthreads();
  { const float* wr = W12 + (long)tid * SDIMN; float a = b12[tid];
    for (int k = 0; k < SDIMN; ++k) a += Bh[k] * wr[k]; Ch[tid] = fmaxf(a, 0.0f) + Ah[tid]; }
  __syncthreads();
  { const float* wr = W21 + (long)tid * SDIMN; float a = b21[tid];
    for (int k = 0; k < SDIMN; ++k) a += Ch[k] * wr[k]; Bh[tid] = fmaxf(a, 0.0f); }
  __syncthreads();
  { const float* wr = W22 + (long)tid * SDIMN; float a = b22[tid];
    for (int k = 0; k < SDIMN; ++k) a += Bh[k] * wr[k]; Ah[tid] = fmaxf(a, 0.0f) + Ch[tid]; }
  __syncthreads();
  if (tid < NTSN) {
    const float* wr = Wo + (long)tid * SDIMN; float a = bo[tid];
    for (int k = 0; k < SDIMN; ++k) a += Ah[k] * wr[k]; ov[tid] = a;
  }
  __syncthreads();
  if (tid == 0) {
    float mx = -INFINITY; for (int j = 0; j < NTSN; ++j) mx = fmaxf(mx, ov[j]);
    float sum = 0.0f;     for (int j = 0; j < NTSN; ++j) sum += expf(ov[j] - mx);
    lse1 = mx + logf(sum);
  }
  __syncthreads();
  if (tid < NTSN) root_scores[b * NTSN + tid] = ov[tid] - lse1;
}

// per-row logsumexp (vocab rows of width ncol)
__global__ __launch_bounds__(256)
void rowlse_kernel(const float* __restrict__ logits, float* __restrict__ lse, int ncol) {
  int row = blockIdx.x;
  const float* lr = logits + (long)row * ncol;
  __shared__ float red[256];
  int tid = threadIdx.x;
  float mx = -INFINITY;
  for (int c = tid; c < ncol; c += 256) mx = fmaxf(mx, lr[c]);
  red[tid] = mx; __syncthreads();
  for (int s = 128; s > 0; s >>= 1) { if (tid < s) red[tid] = fmaxf(red[tid], red[tid + s]); __syncthreads(); }
  mx = red[0]; __syncthreads();
  float sum = 0.0f;
  for (int c = tid; c < ncol; c += 256) sum += expf(lr[c] - mx);
  red[tid] = sum; __syncthreads();
  for (int s = 128; s > 0; s >>= 1) { if (tid < s) red[tid] += red[tid + s]; __syncthreads(); }
  if (tid == 0) lse[row] = mx + logf(red[0]);
}

__global__ void unary_kernel(const float* __restrict__ logits, const float* __restrict__ lse,
                             const int* __restrict__ x, float* __restrict__ unary) {
  int idx = blockIdx.x * 256 + threadIdx.x;
  if (idx >= BBN * NNN * TSN) return;
  int tst = idx % TSN;
  int bp = idx / TSN;
  int pos = bp % NNN;
  int b = bp / NNN;
  int tok = x[b * NNN + pos];
  int row = b * TSN + tst;
  unary[idx] = logits[(long)row * VOCABN + tok] - lse[row];
}

// rexp = exp(logit - rowmax) ; rmaxv = max(log_softmax) = -log(sum exp(l - max))
__global__ __launch_bounds__(256)
void rule_softexp_kernel(const float* __restrict__ logits, float* __restrict__ rexp,
                         float* __restrict__ rmaxv) {
  int row = blockIdx.x;
  const float* lr = logits + (long)row * (SALLN * SALLN);
  float* er = rexp + (long)row * (SALLN * SALLN);
  __shared__ float red[256];
  int tid = threadIdx.x;
  float mx = -INFINITY;
  for (int c = tid; c < SALLN * SALLN; c += 256) mx = fmaxf(mx, lr[c]);
  red[tid] = mx; __syncthreads();
  for (int s = 128; s > 0; s >>= 1) { if (tid < s) red[tid] = fmaxf(red[tid], red[tid + s]); __syncthreads(); }
  mx = red[0]; __syncthreads();
  float sum = 0.0f;
  for (int c = tid; c < SALLN * SALLN; c += 256) { float e = expf(lr[c] - mx); er[c] = e; sum += e; }
  red[tid] = sum; __syncthreads();
  for (int s = 128; s > 0; s >>= 1) { if (tid < s) red[tid] += red[tid + s]; __syncthreads(); }
  if (tid == 0) rmaxv[row] = -logf(red[0]);
}

__global__ void beta_diag_kernel(const float* __restrict__ unary, float* __restrict__ beta) {
  int idx = blockIdx.x * 256 + threadIdx.x;
  if (idx >= BBN * NNN * TSN) return;
  int tst = idx % TSN;
  int bp = idx / TSN;
  int pos = bp % NNN;
  int b = bp / NNN;
  beta[(((long)(b * NNN + pos)) * NNN + pos) * SALLN + NTSN + tst] = unary[idx];
}

// cexp rows: outer-product exp(left-maxl) x exp(right-maxr); cmax = maxl+maxr
__global__ __launch_bounds__(128)
void build_cexp_kernel(const float* __restrict__ beta, float* __restrict__ cexp,
                       float* __restrict__ cmax, int w, int m) {
  __shared__ float el[SALLN], er[SALLN];
  __shared__ float mx2[2];
  int idx = blockIdx.x;
  int mw = m * w;
  int b = idx / mw;
  int rem = idx - b * mw;
  int i = rem / w;
  int s = rem - i * w;
  const float* L = beta + (((long)(b * NNN + i)) * NNN + (i + s)) * SALLN;
  const float* R = beta + (((long)(b * NNN + i + s + 1)) * NNN + (i + w)) * SALLN;
  int tid = threadIdx.x;
  if (tid < SALLN) { el[tid] = L[tid]; er[tid] = R[tid]; }
  __syncthreads();
  if (tid == 0) {
    float ml = -INFINITY, mr = -INFINITY;
    for (int c = 0; c < SALLN; ++c) { ml = fmaxf(ml, el[c]); mr = fmaxf(mr, er[c]); }
    mx2[0] = ml; mx2[1] = mr;
  }
  __syncthreads();
  if (tid < SALLN) { el[tid] = expf(el[tid] - mx2[0]); er[tid] = expf(er[tid] - mx2[1]); }
  __syncthreads();
  float* out = cexp + (long)idx * (SALLN * SALLN);
  for (int c = tid; c < SALLN * SALLN; c += 128) {
    int c1 = c / SALLN;
    int c2 = c - c1 * SALLN;
    out[c] = el[c1] * er[c2];
  }
  if (tid == 0) cmax[idx] = mx2[0] + mx2[1];
}

__global__ __launch_bounds__(32)
void inside_lse_kernel(const float* __restrict__ prod, const float* __restrict__ cmax,
                       const float* __restrict__ rmaxv, float* __restrict__ beta,
                       int w, int m) {
  int bi = blockIdx.x;
  int b = bi / m;
  int i = bi - b * m;
  int a = threadIdx.x;
  if (a >= NTSN) return;
  int mw = m * w;
  float rm = rmaxv[b * NTSN + a];
  float mx = -INFINITY;
  for (int s = 0; s < w; ++s) {
    int row = b * mw + i * w + s;
    float sc = logf(prod[(long)row * NTSN + a] + 1e-38f) + cmax[row] + rm;
    mx = fmaxf(mx, sc);
  }
  float sum = 0.0f;
  for (int s = 0; s < w; ++s) {
    int row = b * mw + i * w + s;
    float sc = logf(prod[(long)row * NTSN + a] + 1e-38f) + cmax[row] + rm;
    sum += expf(sc - mx);
  }
  beta[(((long)(b * NNN + i)) * NNN + (i + w)) * SALLN + a] = mx + logf(sum);
}

__global__ __launch_bounds__(32)
void final_kernel(const float* __restrict__ beta, const float* __restrict__ root_scores,
                  const float* __restrict__ kl, float* __restrict__ out) {
  __shared__ float sh[32];
  int b = blockIdx.x, a = threadIdx.x;
  float v = (a < NTSN)
      ? beta[(((long)(b * NNN)) * NNN + (NNN - 1)) * SALLN + a] + root_scores[b * NTSN + a]
      : -INFINITY;
  sh[a] = v;
  __syncthreads();
  if (a == 0) {
    float mx = -INFINITY; for (int j = 0; j < NTSN; ++j) mx = fmaxf(mx, sh[j]);
    float sum = 0.0f;     for (int j = 0; j < NTSN; ++j) sum += expf(sh[j] - mx);
    out[b] = -(mx + logf(sum));
    out[BBN + b] = kl[b];
  }
}

// ---------------------------------------------------------------------------
static void launch_gemm(hipStream_t stream, const float* A, const float* Wt,
                        const float* bias, const float* resid, float* C,
                        int M, int N, int K, int relu,
                        long sA, long sW, long sC, long sR, int batch) {
  dim3 g((N + 31) / 32, (M + 31) / 32, batch);
  wmma_gemm_kernel<<<g, dim3(32), 0, stream>>>(A, Wt, bias, resid, C, M, N, K, sA, sW, sC, sR, relu);
}

extern "C" void kernel_launch(void* const* d_in, const int* in_sizes, int n_in,
                              void* d_out, int out_size, void* d_ws, size_t ws_size,
                              hipStream_t stream) {
  // setup_inputs() dict order:
  const float* t_emb    = (const float*)d_in[0];
  const float* nt_emb   = (const float*)d_in[1];
  const float* root_emb = (const float*)d_in[2];
  const float* rule_W   = (const float*)d_in[3];
  const float* rule_b   = (const float*)d_in[4];
  const float* r_W0  = (const float*)d_in[5];
  const float* r_b0  = (const float*)d_in[6];
  const float* r_W11 = (const float*)d_in[7];
  const float* r_b11 = (const float*)d_in[8];
  const float* r_W12 = (const float*)d_in[9];
  const float* r_b12 = (const float*)d_in[10];
  const float* r_W21 = (const float*)d_in[11];
  const float* r_b21 = (const float*)d_in[12];
  const float* r_W22 = (const float*)d_in[13];
  const float* r_b22 = (const float*)d_in[14];
  const float* r_Wo  = (const float*)d_in[15];
  const float* r_bo  = (const float*)d_in[16];
  const float* v_W0  = (const float*)d_in[17];
  const float* v_b0  = (const float*)d_in[18];
  const float* v_W11 = (const float*)d_in[19];
  const float* v_b11 = (const float*)d_in[20];
  const float* v_W12 = (const float*)d_in[21];
  const float* v_b12 = (const float*)d_in[22];
  const float* v_W21 = (const float*)d_in[23];
  const float* v_b21 = (const float*)d_in[24];
  const float* v_W22 = (const float*)d_in[25];
  const float* v_b22 = (const float*)d_in[26];
  const float* v_Wo  = (const float*)d_in[27];
  const float* v_bo  = (const float*)d_in[28];
  const float* enc_emb = (const float*)d_in[29];
  const float* Wih_f = (const float*)d_in[30];
  const float* Whh_f = (const float*)d_in[31];
  const float* bih_f = (const float*)d_in[32];
  const float* bhh_f = (const float*)d_in[33];
  const float* Wih_b = (const float*)d_in[34];
  const float* Whh_b = (const float*)d_in[35];
  const float* bih_b = (const float*)d_in[36];
  const float* bhh_b = (const float*)d_in[37];
  const float* ep_W  = (const float*)d_in[38];
  const float* ep_b  = (const float*)d_in[39];
  const int*   xi    = (const int*)d_in[40];
  const float* eps   = (const float*)d_in[41];

  float* W = (float*)d_ws;
  // workspace offsets (floats), all 16B aligned
  const long o_emb   = 0;        // 96*512
  const long o_preF  = 49152;    // 96*2048
  const long o_preB  = 245760;   // 96*2048
  const long o_state = 442368;   // hF,cF,hB,cB : 4*2048
  const long o_hsF   = 450560;   // 96*512
  const long o_hsB   = 499712;   // 96*512
  const long o_hmax  = 548864;   // 4*1024
  const long o_z     = 552960;   // 4*64
  const long o_kl    = 553216;   // 4
  const long o_root  = 553220;   // 4*30
  const long o_tin   = 553344;   // 240*320
  const long o_h0    = 630144;   // 240*256
  const long o_t1    = 691584;   // 240*256
  const long o_h1    = 753024;   // 240*256
  const long o_vlog  = 814464;   // 240*10000
  const long o_vlse  = 3214464;  // 240
  const long o_unary = 3214704;  // 4*24*60
  const long o_ntin  = 3220464;  // 120*320
  const long o_rlog  = 3258864;  // 120*8100
  const long o_rexp  = 4230864;  // 120*8100
  const long o_rmax  = 5202864;  // 120
  const long o_beta  = 5202984;  // 4*24*24*90
  const long o_cexp  = 5410344;  // 4*144*8100 (max)
  const long o_cmax  = 10075944; // 4*144
  const long o_prod  = 10076520; // 4*144*30

  // 1) init LSTM state, gather embeddings, input projections (WMMA)
  fill_kernel<<<(8192 + 255) / 256, 256, 0, stream>>>(W + o_state, 8192, 0.0f);
  emb_gather_kernel<<<(49152 + 255) / 256, 256, 0, stream>>>(xi, enc_emb, W + o_emb);
  launch_gemm(stream, W + o_emb, Wih_f, bih_f, nullptr, W + o_preF, 96, 2048, 512, 0, 0, 0, 0, 0, 1);
  launch_gemm(stream, W + o_emb, Wih_b, bih_b, nullptr, W + o_preB, 96, 2048, 512, 0, 0, 0, 0, 0, 1);

  // 2) BiLSTM scan (24 sequential steps; fwd & bwd in parallel blocks)
  for (int t = 0; t < NNN; ++t) {
    lstm_step_kernel<<<2, 256, 0, stream>>>(W + o_preF, W + o_preB, Whh_f, Whh_b, bhh_f, bhh_b,
                                            W + o_state, W + o_state + 2048,
                                            W + o_state + 4096, W + o_state + 6144,
                                            W + o_hsF, W + o_hsB, t);
  }
  hmax_kernel<<<BBN, 1024, 0, stream>>>(W + o_hsF, W + o_hsB, W + o_hmax);
  ep_kernel<<<BBN, 128, 0, stream>>>(W + o_hmax, ep_W, ep_b, eps, W + o_z, W + o_kl);

  // 3) root MLP + inputs for vocab/rule heads
  root_mlp_kernel<<<BBN, 256, 0, stream>>>(root_emb, W + o_z, r_W0, r_b0, r_W11, r_b11, r_W12, r_b12,
                                           r_W21, r_b21, r_W22, r_b22, r_Wo, r_bo, W + o_root);
  cat_in_kernel<<<(BBN * TSN * 320 + 255) / 256, 256, 0, stream>>>(t_emb, W + o_z, W + o_tin, TSN);
  cat_in_kernel<<<(BBN * NTSN * 320 + 255) / 256, 256, 0, stream>>>(nt_emb, W + o_z, W + o_ntin, NTSN);

  // 4) vocab MLP (all WMMA GEMMs) + log_softmax + unary gather
  launch_gemm(stream, W + o_tin, v_W0, v_b0, nullptr, W + o_h0, 240, 256, 320, 0, 0, 0, 0, 0, 1);
  launch_gemm(stream, W + o_h0, v_W11, v_b11, nullptr, W + o_t1, 240, 256, 256, 1, 0, 0, 0, 0, 1);
  launch_gemm(stream, W + o_t1, v_W12, v_b12, W + o_h0, W + o_h1, 240, 256, 256, 1, 0, 0, 0, 0, 1);
  launch_gemm(stream, W + o_h1, v_W21, v_b21, nullptr, W + o_t1, 240, 256, 256, 1, 0, 0, 0, 0, 1);
  launch_gemm(stream, W + o_t1, v_W22, v_b22, W + o_h1, W + o_h0, 240, 256, 256, 1, 0, 0, 0, 0, 1);
  launch_gemm(stream, W + o_h0, v_Wo, v_bo, nullptr, W + o_vlog, 240, VOCABN, 256, 0, 0, 0, 0, 0, 1);
  rowlse_kernel<<<240, 256, 0, stream>>>(W + o_vlog, W + o_vlse, VOCABN);
  unary_kernel<<<(BBN * NNN * TSN + 255) / 256, 256, 0, stream>>>(W + o_vlog, W + o_vlse, xi, W + o_unary);

  // 5) rule scores (WMMA GEMM) + exp-normalized form for inside
  launch_gemm(stream, W + o_ntin, rule_W, rule_b, nullptr, W + o_rlog, 120, SALLN * SALLN, 320, 0, 0, 0, 0, 0, 1);
  rule_softexp_kernel<<<120, 256, 0, stream>>>(W + o_rlog, W + o_rexp, W + o_rmax);

  // 6) inside algorithm
  fill_kernel<<<(207360 + 255) / 256, 256, 0, stream>>>(W + o_beta, 207360, NEGV);
  beta_diag_kernel<<<(BBN * NNN * TSN + 255) / 256, 256, 0, stream>>>(W + o_unary, W + o_beta);
  for (int w = 1; w < NNN; ++w) {
    int m = NNN - w;
    int mw = m * w;
    build_cexp_kernel<<<BBN * mw, 128, 0, stream>>>(W + o_beta, W + o_cexp, W + o_cmax, w, m);
    launch_gemm(stream, W + o_cexp, W + o_rexp, nullptr, nullptr, W + o_prod,
                mw, NTSN, SALLN * SALLN, 0,
                (long)mw * SALLN * SALLN, (long)NTSN * SALLN * SALLN, (long)mw * NTSN, 0, BBN);
    inside_lse_kernel<<<BBN * m, 32, 0, stream>>>(W + o_prod, W + o_cmax, W + o_rmax, W + o_beta, w, m);
  }

  // 7) -log_Z and kl -> d_out[0:4], d_out[4:8]
  final_kernel<<<BBN, 32, 0, stream>>>(W + o_beta, W + o_root, W + o_kl, (float*)d_out);
  (void)in_sizes; (void)n_in; (void)out_size; (void)ws_size;
}